// IzhikevichNeuronTorch_68032281968902
// MI455X (gfx1250) — compile-verified
//
#include <hip/hip_runtime.h>
#include <stdint.h>

// Izhikevich scan over 2000 steps x 32768 neurons. Pure streaming problem:
// 262MB in + 524MB out @ 23.3 TB/s => ~34us floor. No matmul -> no WMMA.
// gfx1250 path: deep async global->LDS pipeline (ASYNCcnt), b128 async ops
// in SADDR form, non-temporal output stores.
//
// Little's law: ~15MB must be in flight to cover ~650ns HBM latency at
// 23.3 TB/s. 1024 waves x PF=7 chunks x 2KB = ~14MB outstanding.

#define BATCH 16
#define STEPS 2000
#define FEAT  2048
#define TPB   128           // 4 wave32s per block
#define FBLK  (FEAT / TPB)  // 16 feature-chunks per batch row
#define CH    16            // time steps per chunk
#define NCH   (STEPS / CH)  // 125 chunks
#define NBUF  8             // LDS ring buffers
#define PF    7             // chunks prefetched ahead (PF < NBUF)
#define BUFB  (CH * TPB * 4) // bytes per ring buffer (8KB)

__device__ __forceinline__ void issue_chunk(
    const float* __restrict__ base,  // uniform: &in[b][chunk*CH][fc*TPB] (lane offset via g0..g3)
    unsigned l0, unsigned l1, unsigned l2, unsigned l3,
    unsigned g0, unsigned g1, unsigned g2, unsigned g3)
{
    unsigned long long sb = (unsigned long long)(uintptr_t)base;
    // SADDR form: mem = SGPR base + 32-bit per-lane VGPR offset.
    // Each lane copies 16B = 4 consecutive features of one step;
    // 32 lanes = 4 steps; 4 issues = 16 steps (the whole chunk).
    asm volatile("global_load_async_to_lds_b128 %0, %1, %2" :: "v"(l0), "v"(g0), "s"(sb) : "memory");
    asm volatile("global_load_async_to_lds_b128 %0, %1, %2" :: "v"(l1), "v"(g1), "s"(sb) : "memory");
    asm volatile("global_load_async_to_lds_b128 %0, %1, %2" :: "v"(l2), "v"(g2), "s"(sb) : "memory");
    asm volatile("global_load_async_to_lds_b128 %0, %1, %2" :: "v"(l3), "v"(g3), "s"(sb) : "memory");
}

__global__ __launch_bounds__(TPB) void izhikevich_scan_kernel(
    const float* __restrict__ in, float* __restrict__ out)
{
    __shared__ float smem[NBUF * CH * TPB];  // 64 KB ring

    const int tid  = threadIdx.x;
    const int b    = blockIdx.x >> 4;            // batch
    const int fc   = blockIdx.x & (FBLK - 1);    // feature chunk
    const int f    = fc * TPB + tid;             // this lane's neuron

    const int lane = tid & 31;
    const int wfb  = (tid >> 5) * 32;            // wave's feature base in block
    const int prow = lane >> 3;                  // step sub-index 0..3
    const int q    = lane & 7;                   // feature-quad 0..7

    // Uniform base of this block's slice for chunk 0 (feature offset is per-lane)
    const float* gslice = in + ((size_t)b * STEPS) * FEAT + fc * TPB;

    // Per-lane 32-bit global byte offsets, loop-invariant (chunk advance is in SADDR)
    const unsigned gl = (unsigned)((prow * FEAT + wfb + q * 4) * 4);
    const unsigned g0 = gl;
    const unsigned g1 = gl + 1u * 4 * FEAT * 4;
    const unsigned g2 = gl + 2u * 4 * FEAT * 4;
    const unsigned g3 = gl + 3u * 4 * FEAT * 4;

    // Per-lane LDS byte offsets (low 32 bits of flat shared ptr == LDS offset)
    const unsigned ldsBase = (unsigned)(uintptr_t)(&smem[0]);
    const unsigned ll = ldsBase + (unsigned)((prow * TPB + wfb + q * 4) * 4);
    const unsigned l0 = ll;
    const unsigned l1 = ll + 1u * 4 * TPB * 4;
    const unsigned l2 = ll + 2u * 4 * TPB * 4;
    const unsigned l3 = ll + 3u * 4 * TPB * 4;

    float* vout = out + ((size_t)b * (STEPS + 1)) * FEAT + f;
    float* sout = vout + (size_t)BATCH * (STEPS + 1) * FEAT;

    float v = -65.0f;
    float u = 0.2f * -65.0f;

    // ---- prime: issue chunks 0..PF-1 (28 async b128 ops outstanding) ----
    #pragma unroll
    for (int cc = 0; cc < PF; ++cc) {
        const unsigned bo = (unsigned)(cc & (NBUF - 1)) * BUFB;
        issue_chunk(gslice + (size_t)cc * CH * FEAT,
                    l0 + bo, l1 + bo, l2 + bo, l3 + bo, g0, g1, g2, g3);
    }

    // ---- steady state: c in [0, NCH-PF) ----
    for (int c = 0; c < NCH - PF; ++c) {
        {   // issue chunk c+PF into ring slot (c+PF)&7
            const int cc = c + PF;
            const unsigned bo = (unsigned)(cc & (NBUF - 1)) * BUFB;
            issue_chunk(gslice + (size_t)cc * CH * FEAT,
                        l0 + bo, l1 + bo, l2 + bo, l3 + bo, g0, g1, g2, g3);
        }
        // chunks c+1..c+PF = 7*4 = 28 ops may remain; in-order retire
        // guarantees chunk c is resident once ASYNCcnt <= 28.
        asm volatile("s_wait_asynccnt 28" ::: "memory");

        const float* buf = &smem[(c & (NBUF - 1)) * CH * TPB];
        #pragma unroll
        for (int p = 0; p < CH; ++p) {
            const int   t  = c * CH + p;
            const float It = buf[p * TPB + tid];

            const bool  fired = (v >= 30.0f);
            const float vvis  = fired ? 30.0f : v;
            __builtin_nontemporal_store(vvis, vout + (size_t)t * FEAT);
            __builtin_nontemporal_store(fired ? 1.0f : 0.0f, sout + (size_t)t * FEAT);

            const float dv = 0.04f * v * v + 5.0f * v + 140.0f - u + It;
            const float du = 0.02f * (0.2f * v - u);
            v = fired ? -65.0f     : (v + 0.05f * dv);
            u = fired ? (u + 8.0f) : (u + 0.05f * du);
        }
    }

    // ---- drain: last PF chunks are all resident after one full wait ----
    asm volatile("s_wait_asynccnt 0" ::: "memory");
    for (int c = NCH - PF; c < NCH; ++c) {
        const float* buf = &smem[(c & (NBUF - 1)) * CH * TPB];
        #pragma unroll
        for (int p = 0; p < CH; ++p) {
            const int   t  = c * CH + p;
            const float It = buf[p * TPB + tid];

            const bool  fired = (v >= 30.0f);
            const float vvis  = fired ? 30.0f : v;
            __builtin_nontemporal_store(vvis, vout + (size_t)t * FEAT);
            __builtin_nontemporal_store(fired ? 1.0f : 0.0f, sout + (size_t)t * FEAT);

            const float dv = 0.04f * v * v + 5.0f * v + 140.0f - u + It;
            const float du = 0.02f * (0.2f * v - u);
            v = fired ? -65.0f     : (v + 0.05f * dv);
            u = fired ? (u + 8.0f) : (u + 0.05f * du);
        }
    }

    // final sample t == STEPS: raw (unclipped) v + threshold flag
    __builtin_nontemporal_store(v, vout + (size_t)STEPS * FEAT);
    __builtin_nontemporal_store((v >= 30.0f) ? 1.0f : 0.0f,
                                sout + (size_t)STEPS * FEAT);
}

extern "C" void kernel_launch(void* const* d_in, const int* in_sizes, int n_in,
                              void* d_out, int out_size, void* d_ws, size_t ws_size,
                              hipStream_t stream) {
    const float* in = (const float*)d_in[0];
    float* out      = (float*)d_out;
    (void)in_sizes; (void)n_in; (void)d_ws; (void)ws_size; (void)out_size;

    dim3 grid(BATCH * FBLK); // 256 blocks
    dim3 block(TPB);         // 128 threads = 4 wave32s
    hipLaunchKernelGGL(izhikevich_scan_kernel, grid, block, 0, stream, in, out);
}